// GraphConvolution_22239340659136
// MI455X (gfx1250) — compile-verified
//
#include <hip/hip_runtime.h>

// GraphConvolution on MI455X (gfx1250, wave32):
//   h = alpha*x  (init)
//   h[dst] += (1-alpha)*edge_w * x[src]   (one wave per edge, f32 atomics)
//   out = h @ W  (WMMA f32 16x16x4, async global->LDS staged A tiles)

typedef float v2f __attribute__((ext_vector_type(2)));
typedef float v4f __attribute__((ext_vector_type(4)));
typedef float v8f __attribute__((ext_vector_type(8)));

#define D 256
#define LDS_PITCH 260   // 256 + 4 pad -> conflict-free ds reads at row stride
                        // (row pitch = 1040 B, still 16B aligned for b128)

// ---------------------------------------------------------------- init: h = alpha * x
__global__ void gc_init_h(const float* __restrict__ x,
                          const float* __restrict__ alpha_p,
                          float* __restrict__ h, int total4) {
    int i = blockIdx.x * blockDim.x + threadIdx.x;
    if (i >= total4) return;
    float a = alpha_p[0];
    v4f v = ((const v4f*)x)[i];
    v4f r = {v.x * a, v.y * a, v.z * a, v.w * a};
    ((v4f*)h)[i] = r;
}

// ---------------------------------------------------------------- SpMM scatter
// One wave32 per edge: lanes stripe the 256 features (2 x float4 per lane,
// 512B coalesced gather), then 8 global_atomic_add_f32 per lane into h[dst].
__global__ void gc_spmm(const float* __restrict__ edge_w,
                        const float* __restrict__ x,
                        const float* __restrict__ alpha_p,
                        const int* __restrict__ esrc,
                        const int* __restrict__ edst,
                        float* __restrict__ h, int E) {
    int wave = (blockIdx.x * blockDim.x + threadIdx.x) >> 5;
    int lane = threadIdx.x & 31;
    if (wave >= E) return;

    int s = esrc[wave];
    int d = edst[wave];
    float w = edge_w[wave] * (1.0f - alpha_p[0]);

    const v4f* xs = (const v4f*)(x + (size_t)s * D);
    float*     hd = h + (size_t)d * D;

#pragma unroll
    for (int c = 0; c < 2; ++c) {
        int idx = c * 32 + lane;          // float4 index within the row
        v4f v = xs[idx];
        int base = idx * 4;
        unsafeAtomicAdd(hd + base + 0, v.x * w);
        unsafeAtomicAdd(hd + base + 1, v.y * w);
        unsafeAtomicAdd(hd + base + 2, v.z * w);
        unsafeAtomicAdd(hd + base + 3, v.w * w);
    }
}

// ---------------------------------------------------------------- GEMM: out = h @ W
// Block = 256 threads = 8 waves; block owns a 16-row M tile (50000/16 = 3125
// blocks exactly). h tile staged in LDS via CDNA5 async global->LDS DMA
// (ASYNCcnt); each wave computes two 16x16 N tiles with V_WMMA_F32_16X16X4_F32,
// reusing the A fragment across both.
__global__ void __launch_bounds__(256) gc_gemm(const float* __restrict__ h,
                                               const float* __restrict__ W,
                                               float* __restrict__ out) {
    __shared__ float sh[16 * LDS_PITCH];

    const int m0  = blockIdx.x * 16;
    const int tid = threadIdx.x;

    // Cooperative tile stage: 16 rows x 64 float4 = 1024 b128 transfers,
    // 4 per thread, direct global -> LDS (no VGPR bounce).
#pragma unroll
    for (int it = 0; it < 4; ++it) {
        int idx4 = tid + it * 256;
        int row  = idx4 >> 6;     // 64 float4 per row
        int col4 = idx4 & 63;
        const float* gp = h + (size_t)(m0 + row) * D + col4 * 4;
        // Low 32 bits of the generic shared-aperture address are the
        // wave-relative LDS byte offset (flat aperture mapping).
        unsigned lp = (unsigned)(uintptr_t)(sh + row * LDS_PITCH + col4 * 4);
        asm volatile("global_load_async_to_lds_b128 %0, %1, off"
                     :: "v"(lp), "v"(gp)
                     : "memory");
    }
    asm volatile("s_wait_asynccnt 0x0" ::: "memory");
    __syncthreads();

    const int wv   = tid >> 5;
    const int lane = tid & 31;
    const int half = lane >> 4;   // 0: lanes 0-15, 1: lanes 16-31
    const int ncol = lane & 15;
    const int rA   = lane & 15;
    const int n0a  = wv * 32;
    const int n0b  = n0a + 16;

    v8f c0 = {};
    v8f c1 = {};

    for (int k = 0; k < D; k += 4) {
        // A fragment (16x4 f32): lanes 0-15 -> K=k,k+1 ; lanes 16-31 -> K=k+2,k+3
        const float* ap = sh + rA * LDS_PITCH + k + 2 * half;
        v2f a; a.x = ap[0]; a.y = ap[1];

        // B fragment (4x16 f32): VGPR0 -> K rows {k, k+2}, VGPR1 -> {k+1, k+3}
        const float* Wk = W + (size_t)(k + 2 * half) * D;
        v2f b0; b0.x = Wk[n0a + ncol]; b0.y = Wk[D + n0a + ncol];
        v2f b1; b1.x = Wk[n0b + ncol]; b1.y = Wk[D + n0b + ncol];

        c0 = __builtin_amdgcn_wmma_f32_16x16x4_f32(false, a, false, b0,
                                                   (short)0, c0, false, false);
        c1 = __builtin_amdgcn_wmma_f32_16x16x4_f32(false, a, false, b1,
                                                   (short)0, c1, false, false);
    }

    // C/D layout: VGPR r, lanes 0-15 -> (M=r, N=lane); lanes 16-31 -> (M=r+8).
#pragma unroll
    for (int r = 0; r < 8; ++r) {
        int row = m0 + r + half * 8;
        out[(size_t)row * D + n0a + ncol] = c0[r];
        out[(size_t)row * D + n0b + ncol] = c1[r];
    }
}

// ---------------------------------------------------------------- launch
extern "C" void kernel_launch(void* const* d_in, const int* in_sizes, int n_in,
                              void* d_out, int out_size, void* d_ws, size_t ws_size,
                              hipStream_t stream) {
    const float* edge_w = (const float*)d_in[0];
    const float* x      = (const float*)d_in[1];
    const float* W      = (const float*)d_in[2];
    const float* alpha  = (const float*)d_in[3];
    const int*   esrc   = (const int*)d_in[4];
    const int*   edst   = (const int*)d_in[5];

    const int E = in_sizes[0];
    const int N = in_sizes[1] / D;

    float* h   = (float*)d_ws;    // N*D floats = 51.2 MB scratch
    float* out = (float*)d_out;

    const int total4 = (N * D) / 4;
    gc_init_h<<<(total4 + 255) / 256, 256, 0, stream>>>(x, alpha, h, total4);

    const int spmm_blocks = (E + 7) / 8;   // 8 waves (edges) per 256-thread block
    gc_spmm<<<spmm_blocks, 256, 0, stream>>>(edge_w, x, alpha, esrc, edst, h, E);

    gc_gemm<<<N / 16, 256, 0, stream>>>(h, W, out);
}